// KNNGraphBuilder_50766513438988
// MI455X (gfx1250) — compile-verified
//
#include <hip/hip_runtime.h>
#include <hip/hip_bf16.h>

// ---------------------------------------------------------------------------
// Problem constants (from reference setup: B=4, N=8192, F=16, M=7)
// ---------------------------------------------------------------------------
#define BSZ     4
#define N_PTS   8192
#define N_UP    2048      // N/4
#define N_DOWN  6144
#define FDIM    16
#define MK      7

// Output layout: concatenated flat, float32, in return order
#define O_EDGES_J  0                         // B*N_UP*M = 57344
#define O_EDGES_I  57344
#define O_NODEPOS  114688                    // B*N*3 = 98304
#define O_CENTER   212992                    // B*N   = 32768
#define O_FEAT     245760                    // B*N*F = 524288
#define O_UP_SRC   770048
#define O_UP_DST   827392
#define O_NEIGH    884736
// total 942080 floats

typedef __attribute__((ext_vector_type(2))) float v2f;
typedef __attribute__((ext_vector_type(8))) float v8f;

#define BIGF 3.4e38f

// ---------------------------------------------------------------------------
// Kernel 1: per-graph FPS (serial 2047-step scan) + stable partition.
// One workgroup of 1024 threads (32 waves) per graph. The whole per-graph
// pos array (96 KB) is cached in dynamic LDS so the serial dependent read
// pos[last] is an LDS broadcast, not a global round-trip. Argmax uses a
// ping-pong LDS buffer + redundant all-wave reduction: ONE barrier per step.
// ---------------------------------------------------------------------------
__global__ __launch_bounds__(1024)
void fps_partition_kernel(const float* __restrict__ pos,
                          float* __restrict__ out,
                          int* __restrict__ ws_up,
                          int* __restrict__ ws_down) {
  const int b   = blockIdx.x;
  const int tid = threadIdx.x;
  const float* p = pos + (size_t)b * N_PTS * 3;

  extern __shared__ float sp[];               // N_PTS*3 floats = 96 KB
  __shared__ unsigned maskbits[N_PTS / 32];   // 8192-bit mask
  __shared__ float wd[2][32];
  __shared__ int   wi[2][32];
  __shared__ int   wtot[32];
  __shared__ int   wexcl[32];

  for (int i = tid; i < N_PTS * 3; i += 1024) sp[i] = p[i];
  for (int i = tid; i < N_PTS / 32; i += 1024) maskbits[i] = 0u;

  const int base = tid * 8;
  float x[8], y[8], z[8], dist[8];
#pragma unroll
  for (int j = 0; j < 8; ++j) {
    x[j] = p[(base + j) * 3 + 0];
    y[j] = p[(base + j) * 3 + 1];
    z[j] = p[(base + j) * 3 + 2];
    dist[j] = BIGF;
  }
  if (tid == 0) maskbits[0] = 1u;   // sample 0 (dgl default start)
  __syncthreads();

  int last = 0;
  const int lane = tid & 31, wv = tid >> 5;
  for (int s = 1; s < N_UP; ++s) {
    const float px = sp[last * 3 + 0];
    const float py = sp[last * 3 + 1];
    const float pz = sp[last * 3 + 2];
    float bd = -1.0f; int bi = 0x7fffffff;
#pragma unroll
    for (int j = 0; j < 8; ++j) {
      const float dx = x[j] - px, dy = y[j] - py, dz = z[j] - pz;
      const float d  = dx * dx + dy * dy + dz * dz;
      const float nd = fminf(dist[j], d);
      dist[j] = nd;
      const int gi = base + j;
      if (nd > bd || (nd == bd && gi < bi)) { bd = nd; bi = gi; }
    }
    // wave32 argmax reduce (lowest index on ties, matching jnp.argmax)
#pragma unroll
    for (int off = 16; off > 0; off >>= 1) {
      const float od = __shfl_xor(bd, off, 32);
      const int   oi = __shfl_xor(bi, off, 32);
      if (od > bd || (od == bd && oi < bi)) { bd = od; bi = oi; }
    }
    const int pb = s & 1;
    if (lane == 0) { wd[pb][wv] = bd; wi[pb][wv] = bi; }
    __syncthreads();
    // every wave redundantly reduces the 32 wave winners -> identical result
    float vd = wd[pb][lane]; int vi = wi[pb][lane];
#pragma unroll
    for (int off = 16; off > 0; off >>= 1) {
      const float od = __shfl_xor(vd, off, 32);
      const int   oi = __shfl_xor(vi, off, 32);
      if (od > vd || (od == vd && oi < vi)) { vd = od; vi = oi; }
    }
    if (tid == 0) maskbits[vi >> 5] |= (1u << (vi & 31));  // read only after final barrier
    last = vi;
  }
  __syncthreads();

  // ---- stable partition: selected (ascending) -> ws_up, rest -> ws_down ----
  int sel[8]; int cnt = 0;
#pragma unroll
  for (int j = 0; j < 8; ++j) {
    const int gi = base + j;
    sel[j] = (maskbits[gi >> 5] >> (gi & 31)) & 1;
    cnt += sel[j];
  }
  int xs = cnt;
#pragma unroll
  for (int off = 1; off < 32; off <<= 1) {
    const int yv = __shfl_up(xs, off, 32);
    if (lane >= off) xs += yv;
  }
  if (lane == 31) wtot[wv] = xs;
  __syncthreads();
  if (tid < 32) {
    const int t = wtot[tid];
    int ss = t;
#pragma unroll
    for (int off = 1; off < 32; off <<= 1) {
      const int yv = __shfl_up(ss, off, 32);
      if (tid >= off) ss += yv;
    }
    wexcl[tid] = ss - t;
  }
  __syncthreads();

  int pre = wexcl[wv] + (xs - cnt);   // #selected strictly before `base`
#pragma unroll
  for (int j = 0; j < 8; ++j) {
    const int gi = base + j;
    out[O_CENTER + (size_t)b * N_PTS + gi] = sel[j] ? 1.0f : 0.0f;
    if (sel[j]) { ws_up[b * N_UP + pre] = gi; ++pre; }
    else        { ws_down[b * N_DOWN + (gi - pre)] = gi; }
  }
}

// ---------------------------------------------------------------------------
// Kernel 2: compact pu/pd into dense float4 arrays: (x, y, z, bits(node_id)).
// One aligned B128 load then yields coords + id in the KNN kernels.
// ---------------------------------------------------------------------------
__global__ __launch_bounds__(256)
void gather_kernel(const float* __restrict__ pos,
                   const int* __restrict__ ws_up,
                   const int* __restrict__ ws_down,
                   float4* __restrict__ pu4, float4* __restrict__ pd4) {
  const int i = blockIdx.x * 256 + threadIdx.x;
  if (i < BSZ * N_UP) {
    const int b = i / N_UP;
    const int g = ws_up[i];
    const float* p = pos + (size_t)b * N_PTS * 3 + (size_t)g * 3;
    float4 v; v.x = p[0]; v.y = p[1]; v.z = p[2]; v.w = __int_as_float(g);
    pu4[i] = v;
  }
  if (i < BSZ * N_DOWN) {
    const int b = i / N_DOWN;
    const int g = ws_down[i];
    const float* p = pos + (size_t)b * N_PTS * 3 + (size_t)g * 3;
    float4 v; v.x = p[0]; v.y = p[1]; v.z = p[2]; v.w = __int_as_float(g);
    pd4[i] = v;
  }
}

// ---------------------------------------------------------------------------
// Kernel 3: trivial copies (node_pos, feat)
// ---------------------------------------------------------------------------
__global__ __launch_bounds__(256)
void copy_kernel(const float* __restrict__ pos, const float* __restrict__ h,
                 float* __restrict__ out) {
  const int i = blockIdx.x * 256 + threadIdx.x;   // over B*N
  if (i >= BSZ * N_PTS) return;
#pragma unroll
  for (int k = 0; k < 3; ++k)
    out[O_NODEPOS + i * 3 + k] = pos[i * 3 + k];
#pragma unroll
  for (int f = 0; f < FDIM; ++f)
    out[O_FEAT + i * FDIM + f] = h[i * FDIM + f];
}

// ---------------------------------------------------------------------------
// Kernel 4: top-7 KNN via V_WMMA_F32_16X16X4_F32, with column tiles
// software-pipelined through LDS using GLOBAL_LOAD_ASYNC_TO_LDS_B128
// (triple buffer, prefetch distance 2, s_wait_asynccnt discipline).
//   UP=0: up->down graph (6144 cols, store global node ids, dists).
//   UP=1: up->up graph   (2048 cols, diag excluded, store up-list positions).
// ---------------------------------------------------------------------------
template <int UP>
__global__ __launch_bounds__(32)
void knn_kernel(const float4* __restrict__ pu4,
                const float4* __restrict__ pd4,
                float* __restrict__ out) {
  const int b    = blockIdx.x >> 7;     // 128 row tiles per graph
  const int tile = blockIdx.x & 127;
  const int lane = threadIdx.x;         // 0..31 (wave32)
  const int rr   = lane & 15;
  const int hi   = lane >> 4;           // 0: lanes 0-15, 1: lanes 16-31
  const int m0   = tile * 16;

  __shared__ float4 colbuf[3][16];      // async triple buffer (768 B)
  __shared__ float  tileD[256];         // 16x16 distance tile
  __shared__ float  nu16[16];           // |a|^2 per row of tile
  __shared__ float  md[32 * MK];        // merge staging
  __shared__ int    mi[32 * MK];

  const float4* cols = UP ? (pu4 + b * N_UP) : (pd4 + b * N_DOWN);
  const int NT = UP ? (N_UP / 16) : (N_DOWN / 16);

  // Row setup. A operand, f32 16x4 layout: lanes 0-15 hold (K0,K1)=(x,y) for
  // M=lane, lanes 16-31 hold (K2,K3)=(z,0) for M=lane-16.
  const float4 rv = pu4[b * N_UP + m0 + rr];
  const float ax = rv.x, ay = rv.y, az = rv.z;
  const int   gu = __float_as_int(rv.w);          // global node id of row
  v2f A; A.x = hi ? az : ax; A.y = hi ? 0.0f : ay;
  if (!hi) nu16[rr] = ax * ax + ay * ay + az * az;
  __syncthreads();

  // Prologue: async-prefetch column tiles 0 and 1 into LDS (ASYNCcnt path).
  if (lane < 16) {
#pragma unroll
    for (int t = 0; t < 2; ++t) {
      const unsigned lws =
          (unsigned)(unsigned long long)(const void*)&colbuf[t][lane];
      const unsigned long long ga =
          (unsigned long long)(const void*)(cols + t * 16 + lane);
      asm volatile("global_load_async_to_lds_b128 %0, %1, off"
                   :: "v"(lws), "v"(ga) : "memory");
    }
  }

  float kd[MK]; int ki[MK];
#pragma unroll
  for (int k = 0; k < MK; ++k) { kd[k] = BIGF; ki[k] = 0; }

  for (int nt = 0; nt < NT; ++nt) {
    const int cur = nt % 3;
    // Tile nt's async load is complete once at most the (nt+1) prefetch is
    // still outstanding (in-order completion). Final tile: drain to 0.
    if (nt + 1 < NT) asm volatile("s_wait_asynccnt 0x1" ::: "memory");
    else             asm volatile("s_wait_asynccnt 0x0" ::: "memory");

    const float4 cv = colbuf[cur][rr];
    const float bx = cv.x, by = cv.y, bz = cv.z;
    // B operand, f32 4x16 layout mirrored: lanes 0-15 -> (K0,K1), 16-31 -> (K2,0)
    v2f Bm; Bm.x = hi ? bz : bx; Bm.y = hi ? 0.0f : by;
    const float ndn = bx * bx + by * by + bz * bz;

    v8f c = {};
    // 8 args: (neg_a, A, neg_b, B, c_mod, C, reuse_a, reuse_b)
    c = __builtin_amdgcn_wmma_f32_16x16x4_f32(
        /*neg_a=*/false, A, /*neg_b=*/false, Bm,
        /*c_mod=*/(short)0, c, /*reuse_a=*/false, /*reuse_b=*/false);

    __syncthreads();                     // prior tileD fully consumed
    // C layout: VGPR r -> rows r (lanes 0-15) and r+8 (lanes 16-31), N = rr
#pragma unroll
    for (int r = 0; r < 8; ++r) {
      const int mrow = r + hi * 8;
      tileD[mrow * 16 + rr] = nu16[mrow] + ndn - 2.0f * c[r];
    }
    __syncthreads();

    // insertion: this lane owns row rr, columns [hi*8, hi*8+8)
#pragma unroll
    for (int k = 0; k < 8; ++k) {
      const int cl = hi * 8 + k;
      const float d = tileD[rr * 16 + cl];
      const int cp  = nt * 16 + cl;
      const bool skip = UP && (cp == (m0 + rr));   // exclude diagonal
      if (!skip && d < kd[MK - 1]) {
        kd[MK - 1] = d;
        ki[MK - 1] = UP ? cp : __float_as_int(colbuf[cur][cl].w);
#pragma unroll
        for (int q = MK - 1; q > 0; --q) {
          if (kd[q] < kd[q - 1]) {
            const float td = kd[q]; kd[q] = kd[q - 1]; kd[q - 1] = td;
            const int   ti = ki[q]; ki[q] = ki[q - 1]; ki[q - 1] = ti;
          }
        }
      }
    }

    // Prefetch tile nt+2 into the buffer consumed at iter nt-1 (safe: its
    // reads completed before this point in program order).
    if (lane < 16 && nt + 2 < NT) {
      const unsigned lws =
          (unsigned)(unsigned long long)(const void*)&colbuf[(nt + 2) % 3][lane];
      const unsigned long long ga =
          (unsigned long long)(const void*)(cols + (nt + 2) * 16 + lane);
      asm volatile("global_load_async_to_lds_b128 %0, %1, off"
                   :: "v"(lws), "v"(ga) : "memory");
    }
  }

  // merge lane L (cols 0-7 half) with lane L+16 (cols 8-15 half)
#pragma unroll
  for (int k = 0; k < MK; ++k) { md[lane * MK + k] = kd[k]; mi[lane * MK + k] = ki[k]; }
  __syncthreads();
  if (!hi) {
#pragma unroll
    for (int k = 0; k < MK; ++k) {
      const float d = md[(lane + 16) * MK + k];
      const int  ix = mi[(lane + 16) * MK + k];
      if (d < kd[MK - 1]) {
        kd[MK - 1] = d; ki[MK - 1] = ix;
#pragma unroll
        for (int q = MK - 1; q > 0; --q) {
          if (kd[q] < kd[q - 1]) {
            const float td = kd[q]; kd[q] = kd[q - 1]; kd[q - 1] = td;
            const int   ti = ki[q]; ki[q] = ki[q - 1]; ki[q - 1] = ti;
          }
        }
      }
    }
    const int row = m0 + lane;                       // position in up list
    const size_t eb = ((size_t)b * N_UP + row) * MK; // edge base
#pragma unroll
    for (int k = 0; k < MK; ++k) {
      if (UP) {
        out[O_UP_SRC + eb + k] = (float)(ki[k] + b * N_UP);
        out[O_UP_DST + eb + k] = (float)(row   + b * N_UP);
      } else {
        out[O_EDGES_J + eb + k] = (float)(ki[k] + b * N_PTS);
        out[O_EDGES_I + eb + k] = (float)(gu    + b * N_PTS);
        out[O_NEIGH   + eb + k] = kd[k];
      }
    }
  }
}

// ---------------------------------------------------------------------------
extern "C" void kernel_launch(void* const* d_in, const int* in_sizes, int n_in,
                              void* d_out, int out_size, void* d_ws, size_t ws_size,
                              hipStream_t stream) {
  const float* pos = (const float*)d_in[0];   // [B,N,3]
  const float* h   = (const float*)d_in[1];   // [B,N,F]
  float* out = (float*)d_out;

  // workspace: up/down index lists + dense float4 point arrays (~640 KB)
  int* ws_up   = (int*)d_ws;                          // B*N_UP ints
  int* ws_down = ws_up + BSZ * N_UP;                  // B*N_DOWN ints
  float4* pu4  = (float4*)(ws_down + BSZ * N_DOWN);   // B*N_UP float4
  float4* pd4  = pu4 + BSZ * N_UP;                    // B*N_DOWN float4

  // 1) FPS + stable partition + center output (one WG per graph, 96 KB LDS)
  fps_partition_kernel<<<BSZ, 1024, N_PTS * 3 * sizeof(float), stream>>>(
      pos, out, ws_up, ws_down);

  // 2) compact pu/pd into dense float4 (coords + id bits)
  gather_kernel<<<(BSZ * N_DOWN + 255) / 256, 256, 0, stream>>>(
      pos, ws_up, ws_down, pu4, pd4);

  // 3) node_pos / feat copies
  copy_kernel<<<(BSZ * N_PTS + 255) / 256, 256, 0, stream>>>(pos, h, out);

  // 4) up->down KNN (edges_j, edges_i, neigh_dist) — WMMA + async pipeline
  knn_kernel<0><<<BSZ * (N_UP / 16), 32, 0, stream>>>(pu4, pd4, out);

  // 5) up->up KNN (up_src, up_dst) — WMMA + async pipeline, diag excluded
  knn_kernel<1><<<BSZ * (N_UP / 16), 32, 0, stream>>>(pu4, pd4, out);
}